// VariationalFFNDynamic_12996571038353
// MI455X (gfx1250) — compile-verified
//
#include <hip/hip_runtime.h>
#include <cstdint>

// ---------------------------------------------------------------------------
// VariationalFFNDynamic for MI455X (gfx1250, wave32, WMMA + async-LDS).
//
// Exploits orthogonality of exp(skew-symmetric) transports:
//   solve(P S P^T + eps I, dmu) == P Pn diag(1/(sq+eps)) Pn^T P^T dmu
// so the per-pair 16x16 linear solve collapses to diagonal reweighting in
// the transported frame, and every remaining dense op is a 16x16x16 f32
// matmul -> 4 chained v_wmma_f32_16x16x4_f32 per matmul.
// Tile operands are staged into LDS with GLOBAL_LOAD_ASYNC_TO_LDS_B128.
// ---------------------------------------------------------------------------

#define EPSV    1e-6f
#define ALPHA   0.01f
#define KAPPA   1.0f
#define LAMBDAB 1.0f
#define NN      320

typedef __attribute__((ext_vector_type(2))) float v2f;
typedef __attribute__((ext_vector_type(8))) float v8f;

// Compiler-level memory barrier: same-wave LDS is in-order on CDNA5 and the
// backend inserts s_wait_dscnt for aliasing DS deps; this only pins ordering.
static __device__ __forceinline__ void wave_mem_fence() {
  asm volatile("" ::: "memory");
}

// Async global->LDS copy of 1KB (per wave): 2 x b128 per lane.
// GVS mode: LDS dest address in the vdst VGPR, global address = SGPR base + VGPR byte offset
static __device__ __forceinline__ void async_copy_1k(uint32_t lds_byte,
                                                     const float* gbase,
                                                     uint32_t gbyte_off) {
  asm volatile(
      "global_load_async_to_lds_b128 %0, %1, %2 offset:0\n\t"
      "global_load_async_to_lds_b128 %0, %1, %2 offset:512"
      :
      : "v"(lds_byte), "v"(gbyte_off), "s"(gbase)
      : "memory");
}
static __device__ __forceinline__ void wait_async0() {
  asm volatile("s_wait_asynccnt 0x0" ::: "memory");
}

// ---- WMMA helpers ----------------------------------------------------------
// C/D layout (f32 16x16): vgpr r, lane l -> row = r + 8*(l>>4), col = l&15.
// A layout (16x4 f32): lane l -> M = l&15; vgpr0: K = 2*(l>>4), vgpr1: K+1.
// B layout (4x16 f32): vgpr0: K = 2*(l>>4), N = l&15; vgpr1: K+1.
static __device__ __forceinline__ v8f wmma4(v2f a, v2f b, v8f c) {
  return __builtin_amdgcn_wmma_f32_16x16x4_f32(false, a, false, b, (short)0, c,
                                               false, false);
}
static __device__ __forceinline__ v2f ldA(const float* M, int k0, int n) {
  v2f a; a.x = M[n * 16 + k0]; a.y = M[n * 16 + k0 + 1]; return a;
}
// transposed-A read (element (m,k) = M[k][m]) -- same indexing as B operand
static __device__ __forceinline__ v2f ldBT(const float* M, int k0, int n) {
  v2f b; b.x = M[k0 * 16 + n]; b.y = M[(k0 + 1) * 16 + n]; return b;
}
// D = A * B, both row-major 16x16 in LDS, result in C layout
static __device__ __forceinline__ v8f mm16_ss(const float* A, const float* B,
                                              int n, int h) {
  v8f c = {0.f, 0.f, 0.f, 0.f, 0.f, 0.f, 0.f, 0.f};
#pragma unroll
  for (int kk = 0; kk < 4; ++kk) {
    int k0 = kk * 4 + 2 * h;
    c = wmma4(ldA(A, k0, n), ldBT(B, k0, n), c);
  }
  return c;
}
// D = A^T * B
static __device__ __forceinline__ v8f mm16_ts(const float* A, const float* B,
                                              int n, int h) {
  v8f c = {0.f, 0.f, 0.f, 0.f, 0.f, 0.f, 0.f, 0.f};
#pragma unroll
  for (int kk = 0; kk < 4; ++kk) {
    int k0 = kk * 4 + 2 * h;
    c = wmma4(ldBT(A, k0, n), ldBT(B, k0, n), c);
  }
  return c;
}

// ---- Kernel 0: batched 16x16 matrix exponentials (Taylor 10 + 4 squarings) -
// One wave per (node, sign): 640 expm's, 13 WMMA-matmuls each. All LDS
// buffers are wave-private -> no hardware barriers needed. Loops are fully
// unrolled so the 1/k factors fold to immediates (no v_div_* chains between
// dependent WMMAs).
__global__ __launch_bounds__(128) void expm_kernel(
    const float* __restrict__ phi, const float* __restrict__ gens,
    float* __restrict__ P, float* __restrict__ Pn) {
  __shared__ __align__(16) float Am[4][256];
  __shared__ __align__(16) float Tm[4][256];
  const int wave = threadIdx.x >> 5;
  const int l = threadIdx.x & 31;
  const int n = l & 15, h = l >> 4;
  const int idx = blockIdx.x * 4 + wave;  // 0..639 (grid exact)
  const int node = idx >> 1;
  const float sgn = (idx & 1) ? -1.0f : 1.0f;
  const float sc = sgn * (1.0f / 16.0f);  // 2^-squarings
  const float p0 = phi[node * 3 + 0] * sc;
  const float p1 = phi[node * 3 + 1] * sc;
  const float p2 = phi[node * 3 + 2] * sc;
  float* A = Am[wave];
  float* T = Tm[wave];
  v8f out;
#pragma unroll
  for (int r = 0; r < 8; ++r) {
    int row = r + 8 * h;
    float a = p0 * gens[row * 16 + n] + p1 * gens[256 + row * 16 + n] +
              p2 * gens[512 + row * 16 + n];
    A[row * 16 + n] = a;
    T[row * 16 + n] = a;                       // term after k=1
    out[r] = ((row == n) ? 1.0f : 0.0f) + a;   // I + term1
  }
#pragma unroll
  for (int k = 2; k <= 10; ++k) {
    wave_mem_fence();
    v8f c = mm16_ss(T, A, n, h);  // term = term @ A
    const float inv = 1.0f / (float)k;  // compile-time constant (unrolled)
#pragma unroll
    for (int r = 0; r < 8; ++r) { c[r] *= inv; out[r] += c[r]; }
    wave_mem_fence();
#pragma unroll
    for (int r = 0; r < 8; ++r) T[(r + 8 * h) * 16 + n] = c[r];
  }
#pragma unroll
  for (int s = 0; s < 4; ++s) {  // squarings
    wave_mem_fence();
#pragma unroll
    for (int r = 0; r < 8; ++r) T[(r + 8 * h) * 16 + n] = out[r];
    wave_mem_fence();
    out = mm16_ss(T, T, n, h);
  }
  float* dst = ((idx & 1) ? Pn : P) + node * 256;
#pragma unroll
  for (int r = 0; r < 8; ++r) dst[(r + 8 * h) * 16 + n] = out[r];
}

// ---- Kernel 1: per-node prep: sq, w = 1/(sq+eps), r_i = P_i^T mu_i --------
__global__ void node_prep(const float* __restrict__ P,
                          const float* __restrict__ mu,
                          const float* __restrict__ sigma,
                          float* __restrict__ rW, float* __restrict__ wW,
                          float* __restrict__ sqW) {
  int t = blockIdx.x * blockDim.x + threadIdx.x;
  if (t >= NN * 16) return;
  int node = t >> 4, k = t & 15;
  float s = fmaxf(sigma[t], EPSV);
  sqW[t] = s;
  wW[t] = 1.0f / (s + EPSV);
  const float* Pp = P + node * 256;
  const float* m = mu + node * 16;
  float acc = 0.f;
#pragma unroll
  for (int q = 0; q < 16; ++q) acc += Pp[q * 16 + k] * m[q];
  rW[t] = acc;
}

// ---- Kernel 2: pair tiles (16 i x 16 j), 48 WMMA matmuls per tile ----------
// wave j': T_j = Pn_j^T R ; t = T - mu_j ; kl ; U = w (.) t ; Y_j = Pn_j U
// wave i': G_i = P_i Z_i where Z_i cols are y_{ij} over j
// Pn_j / P_i tiles staged via GLOBAL_LOAD_ASYNC_TO_LDS_B128.
__global__ __launch_bounds__(512) void pair_kernel(
    const float* __restrict__ P, const float* __restrict__ Pn,
    const float* __restrict__ rW, const float* __restrict__ wW,
    const float* __restrict__ mu, float* __restrict__ gOut,
    float* __restrict__ klOut) {
  __shared__ __align__(16) float Rb[256];
  __shared__ __align__(16) float Mb[16][256];
  __shared__ __align__(16) float Ub[16][256];
  __shared__ __align__(16) float Yb[16][256];
  const int tid = threadIdx.x;
  const int wave = tid >> 5;
  const int l = tid & 31;
  const int n = l & 15, h = l >> 4;
  const int I0 = blockIdx.x * 16, J0 = blockIdx.y * 16;

  // async copy Pn_{J0+wave} (1KB) into this wave's Mb slot
  const uint32_t ldsMb =
      (uint32_t)(uintptr_t)(&Mb[wave][0]) + (uint32_t)(l * 16);
  async_copy_1k(ldsMb, Pn, (uint32_t)((J0 + wave) * 1024 + l * 16));

  if (tid < 256) {  // R: column ic = r_{I0+ic} (transposed gather)
    int k = tid >> 4, ic = tid & 15;
    Rb[k * 16 + ic] = rW[(I0 + ic) * 16 + k];
  }
  __builtin_prefetch(P + (size_t)(I0 + wave) * 256, 0, 1);  // phase-2 tile
  wait_async0();
  __syncthreads();

  const int j = J0 + wave;
  v8f T = mm16_ts(Mb[wave], Rb, n, h);  // rows = k, cols = i'
  float pk = 0.f;
#pragma unroll
  for (int r = 0; r < 8; ++r) {
    int row = r + 8 * h;
    float t = T[r] - mu[j * 16 + row];
    float wv = wW[j * 16 + row];
    pk += wv * t * t;
    Ub[wave][row * 16 + n] = wv * t;
  }
  float pko = __shfl_xor(pk, 16);
  if (l < 16) klOut[(size_t)(I0 + l) * NN + j] = 0.5f * (pk + pko);
  wave_mem_fence();
  v8f Y = mm16_ss(Mb[wave], Ub[wave], n, h);  // rows = m, cols = i'
#pragma unroll
  for (int r = 0; r < 8; ++r) Yb[wave][(r + 8 * h) * 16 + n] = Y[r];
  __syncthreads();

  // phase 2: async copy P_{I0+wave} into the (now free) Mb slot
  async_copy_1k(ldsMb, P, (uint32_t)((I0 + wave) * 1024 + l * 16));
  wait_async0();
  v8f G = {0.f, 0.f, 0.f, 0.f, 0.f, 0.f, 0.f, 0.f};
#pragma unroll
  for (int kk = 0; kk < 4; ++kk) {
    int k0 = kk * 4 + 2 * h;
    v2f a = ldA(Mb[wave], k0, n);
    v2f b;  // Z_i element (m, jcol=n) = Yb[n][m*16 + i'=wave]
    b.x = Yb[n][k0 * 16 + wave];
    b.y = Yb[n][(k0 + 1) * 16 + wave];
    G = wmma4(a, b, G);
  }
  float* gp = gOut + ((size_t)(I0 + wave) * NN + J0) * 16;
#pragma unroll
  for (int r = 0; r < 8; ++r) gp[n * 16 + (r + 8 * h)] = G[r];
}

// ---- Kernel 3: row softmax (dynamic beta) + alignment reductions + update --
__global__ __launch_bounds__(128) void reduce_update(
    const float* __restrict__ g, const float* __restrict__ kl,
    const float* __restrict__ beta_in, const float* __restrict__ mu_prior,
    const float* __restrict__ sqW, float* __restrict__ muW,
    float* __restrict__ sgW, float* __restrict__ outF,
    const float* __restrict__ lrp, int dynamic_beta) {
  __shared__ float sh[34 * 128];
  const int i = blockIdx.x, tid = threadIdx.x;
  const float* klrow = kl + (size_t)i * NN;
  float mx = 0.f;
  if (dynamic_beta) {
    float m = -3.0e38f;
    for (int jj = tid; jj < NN; jj += 128)
      m = fmaxf(m, -klrow[jj] * (1.0f / KAPPA));
    sh[tid] = m; __syncthreads();
    for (int off = 64; off > 0; off >>= 1) {
      if (tid < off) sh[tid] = fmaxf(sh[tid], sh[tid + off]);
      __syncthreads();
    }
    mx = sh[0];
    __syncthreads();
  }
  float s1[16], s2[16];
#pragma unroll
  for (int k = 0; k < 16; ++k) { s1[k] = 0.f; s2[k] = 0.f; }
  float Z = 0.f, s3 = 0.f;
  for (int jj = tid; jj < NN; jj += 128) {
    float klv = klrow[jj];
    float bw = dynamic_beta ? __expf(-klv * (1.0f / KAPPA) - mx)
                            : beta_in[(size_t)i * NN + jj];
    float kb = klv * bw;
    Z += bw; s3 += kb;
    const float* gp = g + ((size_t)i * NN + jj) * 16;
#pragma unroll
    for (int k = 0; k < 16; ++k) {
      float gv = gp[k];
      s1[k] += bw * gv;
      s2[k] += kb * gv;
    }
  }
#pragma unroll
  for (int k = 0; k < 16; ++k) {
    sh[k * 128 + tid] = s1[k];
    sh[(16 + k) * 128 + tid] = s2[k];
  }
  sh[32 * 128 + tid] = Z;
  sh[33 * 128 + tid] = s3;
  __syncthreads();
  for (int off = 64; off > 0; off >>= 1) {
    if (tid < off) {
#pragma unroll
      for (int row = 0; row < 34; ++row)
        sh[row * 128 + tid] += sh[row * 128 + tid + off];
    }
    __syncthreads();
  }
  const float lr = lrp[0];
  if (tid < 16) {
    int k = tid;
    float invZ = dynamic_beta ? (1.0f / sh[32 * 128]) : 1.0f;
    float avg = sh[k * 128] * invZ;
    float s2n = sh[(16 + k) * 128] * invZ;
    float s3n = sh[33 * 128] * invZ;
    float align = avg + (s2n - s3n * avg) * (1.0f / KAPPA);
    float m = muW[i * 16 + k];
    float grad = ALPHA * (m - mu_prior[i * 16 + k]) + LAMBDAB * align;
    float mn = m - lr * (sqW[i * 16 + k] * grad);
    muW[i * 16 + k] = mn;
    outF[i * 16 + k] = mn;
  } else if (tid < 32) {
    int k = tid - 16;
    float sqv = sqW[i * 16 + k];
    float gs = 0.5f * ALPHA * (1.0f - 1.0f / sqv);
    float ns = 0.5f * sqv * sqv * gs;
    sgW[i * 16 + k] = fmaxf(sgW[i * 16 + k] - lr * ns, EPSV);
  }
}

// ---------------------------------------------------------------------------
extern "C" void kernel_launch(void* const* d_in, const int* in_sizes, int n_in,
                              void* d_out, int out_size, void* d_ws,
                              size_t ws_size, hipStream_t stream) {
  const float* mu_in = (const float*)d_in[0];
  const float* beta_in = (const float*)d_in[1];
  const float* mu_pr = (const float*)d_in[2];
  const float* phi = (const float*)d_in[3];
  const float* sig_in = (const float*)d_in[4];
  const float* gens = (const float*)d_in[5];
  const float* lrp = (const float*)d_in[6];
  float* ws = (float*)d_ws;
  float* P = ws;                       // 320*256
  float* Pn = P + 320 * 256;           // 320*256
  float* muW = Pn + 320 * 256;         // 5120
  float* sgW = muW + 5120;             // 5120
  float* sqW = sgW + 5120;             // 5120
  float* wW = sqW + 5120;              // 5120
  float* rW = wW + 5120;               // 5120
  float* klW = rW + 5120;              // 320*320
  float* gW = klW + 320 * 320;         // 320*320*16  (~7.7 MB total)

  hipMemcpyAsync(muW, mu_in, 5120 * sizeof(float), hipMemcpyDeviceToDevice,
                 stream);
  hipMemcpyAsync(sgW, sig_in, 5120 * sizeof(float), hipMemcpyDeviceToDevice,
                 stream);
  expm_kernel<<<160, 128, 0, stream>>>(phi, gens, P, Pn);
  for (int it = 0; it < 2; ++it) {
    node_prep<<<20, 256, 0, stream>>>(P, muW, sgW, rW, wW, sqW);
    pair_kernel<<<dim3(20, 20), 512, 0, stream>>>(P, Pn, rW, wW, muW, gW, klW);
    reduce_update<<<320, 128, 0, stream>>>(gW, klW, beta_in, mu_pr, sqW, muW,
                                           sgW, (float*)d_out, lrp,
                                           it > 0 ? 1 : 0);
  }
}